// JBU_73212012528102
// MI455X (gfx1250) — compile-verified
//
#include <hip/hip_runtime.h>
#include <cstdint>
#include <cstddef>

// ---------------- problem constants (from reference) ----------------
#define B_   2
#define SC_  384
#define SH_  112
#define SW_  112
#define GC_  3
#define GH_  224
#define GW_  224
#define RAD  3
#define DIA  7
#define NP   49          // 7*7 taps
#define HW_  (GH_*GW_)

typedef __attribute__((ext_vector_type(2))) float v2f;
typedef __attribute__((ext_vector_type(8))) float v8f;

__device__ __forceinline__ int reflect_i(int t, int n) {
  if (t < 0) t = -t;
  if (t >= n) t = 2 * n - 2 - t;
  return t;
}
__device__ __forceinline__ int imax(int a, int b) { return a > b ? a : b; }
__device__ __forceinline__ int imin(int a, int b) { return a < b ? a : b; }

// ---------------- kernel A: channel-wise L2 normalize guidance ----------------
__global__ void gnorm_kernel(const float* __restrict__ g, float* __restrict__ gn) {
  int idx = blockIdx.x * blockDim.x + threadIdx.x;
  if (idx >= B_ * HW_) return;
  int b = idx / HW_, pix = idx - b * HW_;
  const float* gp = g + (size_t)b * GC_ * HW_ + pix;
  float v0 = gp[0], v1 = gp[HW_], v2 = gp[2 * HW_];
  float nrm = sqrtf(v0 * v0 + v1 * v1 + v2 * v2);
  float inv = 1.f / fmaxf(nrm, 1e-12f);
  float* o = gn + (size_t)b * GC_ * HW_ + pix;
  o[0] = v0 * inv; o[HW_] = v1 * inv; o[2 * HW_] = v2 * inv;
}

// ---------------- kernel B: combined normalized bilateral weights ----------------
// ck[b][p][h][w] = ks[p]*rk[b,p,h,w] / max(sum_p, 1e-7)
__global__ void weight_kernel(const float* __restrict__ gn, float* __restrict__ ck,
                              const float* __restrict__ pss, const float* __restrict__ psr) {
  int idx = blockIdx.x * blockDim.x + threadIdx.x;
  if (idx >= B_ * HW_) return;
  int b = idx / HW_, pix = idx - b * HW_;
  int h = pix / GW_, w = pix - h * GW_;

  float ss = fabsf(*pss), sr = fabsf(*psr);
  float i2ss = 1.f / (2.f * ss * ss);
  float i2sr = 1.f / (2.f * sr * sr);

  const float* gb = gn + (size_t)b * GC_ * HW_;
  float c0 = gb[pix], c1 = gb[HW_ + pix], c2 = gb[2 * HW_ + pix];

  float wv[NP];
  float sum = 0.f;
#pragma unroll
  for (int i = 0; i < DIA; ++i) {
    int ny = reflect_i(h + i - RAD, GH_);
#pragma unroll
    for (int j = 0; j < DIA; ++j) {
      int nx = reflect_i(w + j - RAD, GW_);
      int np = ny * GW_ + nx;
      float dot = gb[np] * c0 + gb[HW_ + np] * c1 + gb[2 * HW_ + np] * c2;
      float r = 2.f - dot;
      float dy = (float)(i - RAD), dx = (float)(j - RAD);
      float kv = expf(-r * r * i2sr) * expf(-(dx * dx + dy * dy) * i2ss);
      wv[i * DIA + j] = kv;
      sum += kv;
    }
  }
  float inv = 1.f / fmaxf(sum, 1e-7f);
  size_t base = (size_t)b * NP * HW_ + (size_t)h * GW_ + w;
#pragma unroll
  for (int p = 0; p < NP; ++p) ck[base + (size_t)p * HW_] = wv[p] * inv;
}

// ---------------- kernel C: banded-matmul adaptive conv via fp32 WMMA ----------------
// Per block: (b, 64-channel group, output row gy, 16-pixel tile x0).
//   D[16ch x 16px] = sum_{i=0..6} A_i[16 x 24] * W_i[24 x 16]
//   A_i[c, x'] = bilinear-upsampled source (reflect-padded) at (gy-3+i, x0-3+x')
//   W_i[x', x] = ck[i*7 + (x'-x), gy, x0+x]  (banded; zeros pre-baked into LDS)
#define ASTRIDE 169   // 7*24=168, +1 pad -> gcd(169,64)=1, conflict-free LDS rows

__global__ void __launch_bounds__(128)
jbu_wmma_kernel(const float* __restrict__ src, const float* __restrict__ ck,
                float* __restrict__ out) {
  __shared__ float sA[64 * ASTRIDE];     // upsampled source tile, [ch][i*24+x']
  __shared__ float sW[NP * 16];          // raw ck tile, [p][x]
  __shared__ float sWB[DIA * 24 * 16];   // banded W, [i][x'][x] (zeros included)

  const int x0 = blockIdx.x * 16;
  const int gy = blockIdx.y;
  const int zz = blockIdx.z;
  const int b  = zz / (SC_ / 64);
  const int cg = zz % (SC_ / 64);
  const int c0 = cg * 64;
  const int t  = threadIdx.x;

  // ---- async-stage ck[b, 0:49, gy, x0:x0+16] into LDS (ASYNCcnt path) ----
  {
    const float* wrow = ck + ((size_t)b * NP * GH_ + gy) * (size_t)GW_ + x0;
    for (int e = t; e < NP * 4; e += 128) {      // 49 rows * 4 x b128
      int p = e >> 2, q = e & 3;
      const float* ga = wrow + (size_t)p * HW_ + q * 4;
      unsigned lds = (unsigned)(uintptr_t)(&sW[p * 16 + q * 4]);
      asm volatile("global_load_async_to_lds_b128 %0, %1, off"
                   :: "v"(lds), "v"((unsigned long long)(uintptr_t)ga)
                   : "memory");
    }
  }

  // ---- stage bilinear 2x upsample (half-pixel, edge-clamp) of source rows ----
  // (overlaps with the async weight DMA above)
  const float* sbase = src + ((size_t)b * SC_ + c0) * (SH_ * SW_);
  for (int e = t; e < 64 * DIA * 24; e += 128) {   // 10752 / 128 = 84 exact
    int c   = e / (DIA * 24);
    int rem = e - c * (DIA * 24);
    int i   = rem / 24;
    int xc  = rem - i * 24;
    int yy = reflect_i(gy - RAD + i, GH_);
    int xx = reflect_i(x0 - RAD + xc, GW_);
    const float* sp = sbase + (size_t)c * (SH_ * SW_);
    // src coord = 0.5*t - 0.25 ; even t -> (t/2-1, w=0.75), odd t -> (t/2, w=0.25)
    int   iy = (yy & 1) ? (yy >> 1) : (yy >> 1) - 1;
    float wy = (yy & 1) ? 0.25f : 0.75f;
    int   ix = (xx & 1) ? (xx >> 1) : (xx >> 1) - 1;
    float wx = (xx & 1) ? 0.25f : 0.75f;
    int iy0 = imax(iy, 0), iy1 = imin(iy + 1, SH_ - 1);
    int ix0 = imax(ix, 0), ix1 = imin(ix + 1, SW_ - 1);
    float s00 = sp[iy0 * SW_ + ix0], s01 = sp[iy0 * SW_ + ix1];
    float s10 = sp[iy1 * SW_ + ix0], s11 = sp[iy1 * SW_ + ix1];
    float tp = s00 + wx * (s01 - s00);
    float bt = s10 + wx * (s11 - s10);
    sA[c * ASTRIDE + rem] = tp + wy * (bt - tp);
  }

  asm volatile("s_wait_asynccnt 0" ::: "memory");
  __syncthreads();

  // ---- expand raw ck tile into the banded 24x16 W matrices (zeros baked in) ----
  // kills the exec-masked predicated loads in the hot loop
  for (int e = t; e < DIA * 24 * 16; e += 128) {   // 2688 / 128 = 21 exact
    int i   = e / (24 * 16);
    int rem = e - i * (24 * 16);
    int xp  = rem >> 4;          // x' in 0..23
    int x   = rem & 15;          // pixel col
    int j   = xp - x;            // band offset
    float v = (j >= 0 && j < DIA) ? sW[(i * DIA + j) * 16 + x] : 0.f;
    sWB[e] = v;
  }
  __syncthreads();

  // ---- per-wave 16x16 output tile via V_WMMA_F32_16X16X4_F32 ----
  const int lane = t & 31;
  const int wid  = t >> 5;          // 4 waves -> 4 channel sub-tiles of 16
  const int m    = lane & 15;       // A row (channel) == B col (pixel)
  const int kh   = lane >> 4;       // K half: lanes 0-15 hold K=0,1; 16-31 hold K=2,3

  v8f acc = {0.f, 0.f, 0.f, 0.f, 0.f, 0.f, 0.f, 0.f};
  const float* aRow = &sA[(wid * 16 + m) * ASTRIDE];
  const float* wCol = &sWB[m];

  for (int i = 0; i < DIA; ++i) {
#pragma unroll
    for (int kk = 0; kk < 6; ++kk) {          // K = 24 per tap-row
      int xp0 = kk * 4 + 2 * kh;              // this lane's K columns xp0, xp0+1
      v2f a, w2;
      a.x = aRow[i * 24 + xp0];
      a.y = aRow[i * 24 + xp0 + 1];
      const float* wB = wCol + (i * 24 + xp0) * 16;
      w2.x = wB[0];                           // W[xp0,   m]
      w2.y = wB[16];                          // W[xp0+1, m]
      acc = __builtin_amdgcn_wmma_f32_16x16x4_f32(
          /*neg_a=*/false, a, /*neg_b=*/false, w2,
          /*c_mod=*/(short)0, acc, /*reuse_a=*/false, /*reuse_b=*/false);
    }
  }

  // ---- write D: VGPR r holds rows M=r (lanes 0-15) and M=8+r (lanes 16-31) ----
  float* obase = out + (((size_t)b * SC_ + c0 + wid * 16) * GH_ + gy) * (size_t)GW_ + x0;
#pragma unroll
  for (int r = 0; r < 8; ++r) {
    int row = r + 8 * kh;
    obase[(size_t)row * HW_ + m] = acc[r];
  }
}

// ---------------- host-side launcher ----------------
extern "C" void kernel_launch(void* const* d_in, const int* in_sizes, int n_in,
                              void* d_out, int out_size, void* d_ws, size_t ws_size,
                              hipStream_t stream) {
  (void)in_sizes; (void)n_in; (void)out_size; (void)ws_size;
  const float* source   = (const float*)d_in[0];
  const float* guidance = (const float*)d_in[1];
  const float* sspat    = (const float*)d_in[2];
  const float* srange   = (const float*)d_in[3];
  float* out = (float*)d_out;

  // workspace: gn (normalized guidance) then ck (normalized weights)
  float* gn = (float*)d_ws;                       // B*GC*GH*GW  = 301,056 f
  float* ck = gn + (size_t)B_ * GC_ * HW_;        // B*49*GH*GW = 4,917,248 f (~21 MB total)

  const int npix = B_ * HW_;
  const int tpb  = 256;
  const int nblk = (npix + tpb - 1) / tpb;

  gnorm_kernel<<<nblk, tpb, 0, stream>>>(guidance, gn);
  weight_kernel<<<nblk, tpb, 0, stream>>>(gn, ck, sspat, srange);

  dim3 grid(GW_ / 16, GH_, B_ * (SC_ / 64));      // (14, 224, 12)
  jbu_wmma_kernel<<<grid, 128, 0, stream>>>(source, ck, out);
}